// WeightedRankNetLoss_56418690400289
// MI455X (gfx1250) — compile-verified
//
#include <hip/hip_runtime.h>

typedef __attribute__((ext_vector_type(2))) float v2f;
typedef __attribute__((ext_vector_type(8))) float v8f;

#define N_CAND 8192
#define JCHUNK 1024                    // columns staged in LDS per block (8 KB)
#define NBLK_X (N_CAND / 128)          // 64  (each block: 8 waves x 16-row strips)
#define NBLK_Y (N_CAND / JCHUNK)       // 8
#define NPART  (NBLK_X * NBLK_Y)       // 512 partial sums

#define LOG2E_F 1.4426950408889634f
#define LN2_F   0.6931471805599453f

// Pass 1: each wave owns a 16-row i-strip; per 16x16 tile, one
// v_wmma_f32_16x16x4_f32 builds the rank-2 difference tile d[i][j]=l_j-l_i,
// then VALU applies stable softplus * mask / (r_i+r_j) and accumulates.
// softplus uses raw v_exp_f32/v_log_f32: arguments are range-restricted
// ( exp2 arg <= 0, log2 arg in (1,2] ) so the slow libm edge handling is
// provably unnecessary.
__global__ __launch_bounds__(256) void ranknet_pair_kernel(
    const float* __restrict__ logits, const int* __restrict__ rankings,
    float* __restrict__ partials)
{
  __shared__ float lds_l[JCHUNK];
  __shared__ float lds_r[JCHUNK];
  __shared__ float lds_wsum[8];

  const int tid   = threadIdx.x;
  const int lane  = tid & 31;
  const int wave  = tid >> 5;
  const int jbase = blockIdx.y * JCHUNK;

  // Stage the j-chunk (logits + ranks as float) into LDS once per block.
  for (int t = tid; t < JCHUNK; t += 256) {
    lds_l[t] = logits[jbase + t];
    lds_r[t] = (float)rankings[jbase + t];
  }
  __syncthreads();

  const int  i0   = (blockIdx.x * 8 + wave) * 16;  // this wave's 16-row strip
  const bool lo   = lane < 16;
  const int  nloc = lane & 15;

  // A-matrix (16x4 f32): lanes 0..15 hold row M=lane with K=0..1; K=2..3 zero.
  // A[m][0] = -l_i, A[m][1] = 1  ->  D = l_j - l_i after multiplying B below.
  const float lrow = logits[i0 + nloc];
  v2f a;
  a.x = lo ? -lrow : 0.0f;
  a.y = lo ?  1.0f : 0.0f;

  // C/D layout rows: m = v + (lane<16 ? 0 : 8). Preload r_i per VGPR slot.
  const int mbase = i0 + (lo ? 0 : 8);
  float rrow[8];
  #pragma unroll
  for (int v = 0; v < 8; ++v) rrow[v] = (float)rankings[mbase + v];

  float acc[8];
  #pragma unroll
  for (int v = 0; v < 8; ++v) acc[v] = 0.0f;

  // Skip tiles entirely below the diagonal: need jbase+16*jt+15 > i0.
  int jt0 = (i0 - 14) - jbase;
  jt0 = jt0 > 0 ? (jt0 + 15) >> 4 : 0;

  for (int jt = jt0; jt < JCHUNK / 16; ++jt) {
    const int   j0    = jbase + jt * 16;
    const int   jglob = j0 + nloc;
    const float lj    = lds_l[jt * 16 + nloc];   // broadcast reads
    const float rj    = lds_r[jt * 16 + nloc];

    // B-matrix (4x16 f32): B[0][n]=1, B[1][n]=l_j, rows 2..3 zero.
    v2f b;
    b.x = lo ? 1.0f : 0.0f;
    b.y = lo ? lj   : 0.0f;

    v8f c = {};
    c = __builtin_amdgcn_wmma_f32_16x16x4_f32(
        /*neg_a=*/false, a, /*neg_b=*/false, b,
        /*c_mod=*/(short)0, c, /*reuse_a=*/false, /*reuse_b=*/false);

    #pragma unroll
    for (int v = 0; v < 8; ++v) {
      const float d  = c[v];                       // l_j - l_i
      const int   ig = mbase + v;
      const float ri = rrow[v];
      const bool  m  = (ig < jglob) & (ri < rj);   // i<j && r_i<r_j
      const float w  = m ? __builtin_amdgcn_rcpf(ri + rj) : 0.0f;
      // stable softplus on raw HW trans ops:
      //   e = 2^(-|d|*log2e) in (0,1];  sp = max(d,0) + ln2*log2(1+e)
      const float e  = __builtin_amdgcn_exp2f(-fabsf(d) * LOG2E_F);
      const float sp = fmaf(LN2_F, __builtin_amdgcn_logf(1.0f + e),
                            fmaxf(d, 0.0f));
      acc[v] = fmaf(w, sp, acc[v]);
    }
  }

  float s = 0.0f;
  #pragma unroll
  for (int v = 0; v < 8; ++v) s += acc[v];
  #pragma unroll
  for (int off = 16; off > 0; off >>= 1) s += __shfl_xor(s, off, 32);

  if (lane == 0) lds_wsum[wave] = s;
  __syncthreads();
  if (tid == 0) {
    float t = 0.0f;
    #pragma unroll
    for (int w = 0; w < 8; ++w) t += lds_wsum[w];
    partials[blockIdx.y * gridDim.x + blockIdx.x] = t;
  }
}

// Pass 2: deterministic reduction of the 512 block partials, scale by 1/N.
__global__ __launch_bounds__(256) void ranknet_reduce_kernel(
    const float* __restrict__ partials, float* __restrict__ out)
{
  __shared__ float lds[8];
  const int tid = threadIdx.x;
  float s = 0.0f;
  for (int i = tid; i < NPART; i += 256) s += partials[i];
  #pragma unroll
  for (int off = 16; off > 0; off >>= 1) s += __shfl_xor(s, off, 32);
  const int lane = tid & 31, wave = tid >> 5;
  if (lane == 0) lds[wave] = s;
  __syncthreads();
  if (tid == 0) {
    float t = 0.0f;
    #pragma unroll
    for (int w = 0; w < 8; ++w) t += lds[w];
    out[0] = t / (float)N_CAND;
  }
}

extern "C" void kernel_launch(void* const* d_in, const int* in_sizes, int n_in,
                              void* d_out, int out_size, void* d_ws, size_t ws_size,
                              hipStream_t stream) {
  const float* logits   = (const float*)d_in[0];
  const int*   rankings = (const int*)d_in[1];
  float*       out      = (float*)d_out;
  float*       partials = (float*)d_ws;   // 512 floats, fully overwritten each call

  dim3 grid(NBLK_X, NBLK_Y);
  ranknet_pair_kernel<<<grid, 256, 0, stream>>>(logits, rankings, partials);
  ranknet_reduce_kernel<<<1, 256, 0, stream>>>(partials, out);
}